// CapsuleLayer_12592844112104
// MI455X (gfx1250) — compile-verified
//
#include <hip/hip_runtime.h>
#include <math.h>

typedef float v2f __attribute__((ext_vector_type(2)));
typedef float v8f __attribute__((ext_vector_type(8)));

#define B_TOT   32
#define BH      16            // batches per block (batch half)
#define NCAPS   64
#define DIM     32
#define INCAPS  2048
#define INDIM   16
#define ICHUNK  16
#define NIB     (INCAPS / ICHUNK)   // 128 i-blocks
#define THREADS 1024
#define WTILE   (DIM * INDIM)       // 512 floats = 2 KB per (n,i)

// dynamic LDS layout (floats)
#define UH_FLOATS  (BH * NCAPS * DIM)          // 32768 (128 KB)
#define LTC_OFF    UH_FLOATS
#define MROW_OFF   (LTC_OFF + BH * NCAPS)      // + 1024
#define ZROW_OFF   (MROW_OFF + BH)
#define SMEM_FLOATS (ZROW_OFF + BH)            // 33824 floats = 135296 B

// XOR swizzle on d keeps per-(b,n) d-rows bank-conflict-free in LDS
__device__ __forceinline__ int sw_idx(int b, int n, int d) {
    return (b * NCAPS + n) * DIM + (d ^ (n & 31));
}

__global__ void __launch_bounds__(THREADS)
caps_round_kernel(const float* __restrict__ x,      // [32][2048][16]
                  const float* __restrict__ W,      // [64][2048][32][16]
                  const float* __restrict__ vT,     // [32d][32b][64n] (prev round)
                  float*       __restrict__ logits, // [32b][64n][2048i]
                  float*       __restrict__ spart,  // [NIB][32b][64n][32d]
                  int r)
{
    extern __shared__ float sm[];
    float* uh   = sm;               // u_hat tile for current i
    float* ltc  = sm + LTC_OFF;     // logits / exp scratch [BH][NCAPS]
    float* mrow = sm + MROW_OFF;    // per-batch max
    float* zrow = sm + ZROW_OFF;    // per-batch 1/sum

    const int bh   = blockIdx.x;    // batch half (fastest-varying: L2 sharing of W)
    const int ib   = blockIdx.y;    // i-chunk
    const int tid  = threadIdx.x;
    const int lane = tid & 31;
    const int wave = tid >> 5;      // 0..31
    const int l16  = lane & 15;
    const int lhi  = lane >> 4;     // 0 or 1
    const int n0   = wave * 2;      // this wave owns capsules n0, n0+1
    const int tb   = tid >> 6;      // routing-phase batch (0..15)
    const int tn   = tid & 63;      // routing-phase capsule

    // Loop-invariant bases (avoid per-iteration spills/reloads)
    const float* xbase = x + (size_t)(bh * BH + l16) * (INCAPS * INDIM) + 2 * lhi
                           + (size_t)ib * ICHUNK * INDIM;
    const float* wbase = W + (size_t)n0 * (INCAPS * WTILE) + (size_t)ib * ICHUNK * WTILE
                           + l16 * INDIM + 2 * lhi;
    const char*  pf0   = (const char*)(W + ((size_t)n0       * INCAPS + ib * ICHUNK) * WTILE) + lane * 64;
    const char*  pf1   = (const char*)(W + ((size_t)(n0 + 1) * INCAPS + ib * ICHUNK) * WTILE) + lane * 64;

    float sreg[DIM];
#pragma unroll
    for (int d = 0; d < DIM; ++d) sreg[d] = 0.f;

    for (int ii = 0; ii < ICHUNK; ++ii) {
        const int i = ib * ICHUNK + ii;

        // ---------- Phase W: u_hat[:, :, i, :] via V_WMMA_F32_16X16X4_F32 ----------
        // A (16x4 f32): lane l: A.x = x[b=l16][K=2*lhi], A.y = x[b=l16][K=2*lhi+1]
        v2f a[4];
        {
            const float* xp = xbase + ii * INDIM;
            a[0] = *(const v2f*)(xp);
            a[1] = *(const v2f*)(xp + 4);
            a[2] = *(const v2f*)(xp + 8);
            a[3] = *(const v2f*)(xp + 12);
        }
        // Preload ALL B fragments: one load clause -> ~16 loads in flight/wave.
        // B (4x16 f32): lane l: B.x = W[d=dh*16+l16][j=4k+2*lhi], B.y = j+1
        v2f bf[16];                 // [nn][dh][k]
        {
            const float* wp0 = wbase + (size_t)ii * WTILE;
#pragma unroll
            for (int nn = 0; nn < 2; ++nn)
#pragma unroll
                for (int dh = 0; dh < 2; ++dh)
#pragma unroll
                    for (int k = 0; k < 4; ++k)
                        bf[(nn * 2 + dh) * 4 + k] =
                            *(const v2f*)(wp0 + (size_t)nn * (INCAPS * WTILE)
                                              + dh * (16 * INDIM) + 4 * k);
        }
        // Prefetch next i's W tiles for this wave's two capsules (2 KB each:
        // 32 lanes x 64B stride covers the whole block). Guarded: stay in-bounds.
        if (ii + 1 < ICHUNK) {
            const size_t pofs = (size_t)(ii + 1) * (WTILE * 4);
            __builtin_prefetch(pf0 + pofs, 0, 3);
            __builtin_prefetch(pf1 + pofs, 0, 3);
        }

#pragma unroll
        for (int nn = 0; nn < 2; ++nn) {
            const int n = n0 + nn;
#pragma unroll
            for (int dh = 0; dh < 2; ++dh) {
                v8f acc = {0.f, 0.f, 0.f, 0.f, 0.f, 0.f, 0.f, 0.f};
#pragma unroll
                for (int k = 0; k < 4; ++k)
                    acc = __builtin_amdgcn_wmma_f32_16x16x4_f32(
                        false, a[k], false, bf[(nn * 2 + dh) * 4 + k],
                        (short)0, acc, false, false);
                // D layout: VGPR kk -> row M = kk + 8*lhi, col N = l16
#pragma unroll
                for (int kk = 0; kk < 8; ++kk)
                    uh[sw_idx(kk + 8 * lhi, n, dh * 16 + l16)] = acc[kk];
            }
        }
        __syncthreads();

        // ---------- Phase T + softmax over n (rounds 1,2) ----------
        float c;
        if (r > 0) {
            float t = 0.f;
#pragma unroll
            for (int d = 0; d < DIM; ++d)
                t += vT[((size_t)d * B_TOT + bh * BH + tb) * NCAPS + tn] *
                     uh[sw_idx(tb, tn, d)];
            const size_t lix = ((size_t)(bh * BH + tb) * NCAPS + tn) * INCAPS + i;
            float lt = t;
            if (r == 2) lt += logits[lix];   // b2 = b1 + t1
            else        logits[lix] = lt;    // store b1 = t0
            ltc[tb * NCAPS + tn] = lt;
            __syncthreads();
            if (tid < BH) {                  // per-batch max over 64 capsules
                float m = -3.4e38f;
                for (int q = 0; q < NCAPS; ++q) m = fmaxf(m, ltc[tid * NCAPS + q]);
                mrow[tid] = m;
            }
            __syncthreads();
            float e = expf(ltc[tb * NCAPS + tn] - mrow[tb]);
            ltc[tb * NCAPS + tn] = e;
            __syncthreads();
            if (tid < BH) {
                float z = 0.f;
                for (int q = 0; q < NCAPS; ++q) z += ltc[tid * NCAPS + q];
                zrow[tid] = 1.f / z;
            }
            __syncthreads();
            c = e * zrow[tb];
        } else {
            c = 1.f / 64.f;                  // softmax of zero logits
        }

        // ---------- Phase S: s[b,n,:] += c * u_hat[b,n,i,:] ----------
#pragma unroll
        for (int d = 0; d < DIM; ++d)
            sreg[d] = fmaf(c, uh[sw_idx(tb, tn, d)], sreg[d]);
        __syncthreads();                     // uh reused next i
    }

    float* sp = spart + (((size_t)ib * B_TOT + bh * BH + tb) * NCAPS + tn) * DIM;
#pragma unroll
    for (int d = 0; d < DIM; ++d) sp[d] = sreg[d];
}

// Deterministic cross-block reduction over i-chunks + squash.
__global__ void caps_reduce_squash(const float* __restrict__ spart,
                                   float* __restrict__ vT_out,   // [32d][32b][64n]
                                   float* __restrict__ out,      // [32b][64n][32d]
                                   int is_final)
{
    const int bn = blockIdx.x;   // b*64 + n
    const int d  = threadIdx.x;  // 0..31
    float acc = 0.f;
    for (int ib = 0; ib < NIB; ++ib)
        acc += spart[((size_t)ib * (B_TOT * NCAPS) + bn) * DIM + d];

    __shared__ float red[DIM + 1];
    red[d] = acc * acc;
    __syncthreads();
    if (d == 0) {
        float s2 = 0.f;
        for (int k = 0; k < DIM; ++k) s2 += red[k];
        red[DIM] = s2;
    }
    __syncthreads();
    const float s2 = red[DIM];
    const float scale = s2 / (1.0f + s2) / sqrtf(s2 + 1e-7f);
    const float v = scale * acc;
    if (is_final) {
        out[(size_t)bn * DIM + d] = v;
    } else {
        const int b = bn >> 6, n = bn & 63;
        vT_out[((size_t)d * B_TOT + b) * NCAPS + n] = v;  // transposed for coalesced reads
    }
}

extern "C" void kernel_launch(void* const* d_in, const int* in_sizes, int n_in,
                              void* d_out, int out_size, void* d_ws, size_t ws_size,
                              hipStream_t stream) {
    const float* x = (const float*)d_in[0];   // [32,2048,16]
    const float* W = (const float*)d_in[1];   // [64,2048,32,16]
    float* out = (float*)d_out;               // [32,64,32]
    char* ws = (char*)d_ws;

    float* vT     = (float*)(ws);                                    // 256 KB
    float* logits = (float*)(ws + (size_t)(1u << 20));               // 16 MB
    float* spart  = (float*)(ws + (size_t)(1u << 20) + (16u << 20)); // 32 MB

    const dim3 grid(2, NIB, 1);   // bh fastest -> W tile shared in L2
    const size_t smem = SMEM_FLOATS * sizeof(float);
    for (int r = 0; r < 3; ++r) {
        caps_round_kernel<<<grid, THREADS, smem, stream>>>(x, W, vT, logits, spart, r);
        caps_reduce_squash<<<B_TOT * NCAPS, DIM, 0, stream>>>(spart, vT, out, r == 2 ? 1 : 0);
    }
}